// MatrixNet_40114994544972
// MI455X (gfx1250) — compile-verified
//
#include <hip/hip_runtime.h>
#include <math.h>

#define M_NODES 4096
#define NOFF 13
#define BLK 16
#define NBLK 256

typedef float v2f __attribute__((ext_vector_type(2)));
typedef float v8f __attribute__((ext_vector_type(8)));

__constant__ int OFF_DR[NOFF] = {-2,-1,-1,-1, 0, 0, 0, 0, 0, 1, 1, 1, 2};
__constant__ int OFF_DC[NOFF] = { 0,-1, 0, 1,-2,-1, 0, 1, 2,-1, 0, 1, 0};
// mirror offset index is (12 - t)

// ---------------------------------------------------------------- scores ----
__device__ __forceinline__ float imval(const float* __restrict__ img, int row, int col) {
  return (row >= 0 && row < 64 && col >= 0 && col < 64) ? img[row * 64 + col] : 0.0f;
}

__device__ __forceinline__ float lenet9(const float* p,
    const float* __restrict__ cw,  const float* __restrict__ cb,
    const float* __restrict__ f1w, const float* __restrict__ f1b,
    const float* __restrict__ f2w, const float* __restrict__ f2b,
    const float* __restrict__ f3w, const float* __restrict__ f3b) {
  float h1[2];
  for (int o = 0; o < 2; ++o) { float s = cb[o];  for (int t = 0; t < 9; ++t) s += p[t]  * cw[o * 9 + t];  h1[o] = s; }
  float h2[9];
  for (int o = 0; o < 9; ++o) { float s = f1b[o]; for (int t = 0; t < 2; ++t) s += h1[t] * f1w[o * 2 + t]; h2[o] = s; }
  float h3[9];
  for (int o = 0; o < 9; ++o) { float s = f2b[o]; for (int t = 0; t < 9; ++t) s += h2[t] * f2w[o * 9 + t]; h3[o] = fmaxf(s, 0.0f); }
  float s4 = f3b[0];
  for (int t = 0; t < 9; ++t) s4 += h3[t] * f3w[t];
  return fmaxf(s4, 0.0f);
}

__global__ __launch_bounds__(256) void k_scores(
    const float* __restrict__ img,
    const float* __restrict__ cw,  const float* __restrict__ cb,
    const float* __restrict__ f1w, const float* __restrict__ f1b,
    const float* __restrict__ f2w, const float* __restrict__ f2b,
    const float* __restrict__ f3w, const float* __restrict__ f3b,
    float* __restrict__ Sdir) {
  int tid = blockIdx.x * 256 + threadIdx.x;
  if (tid >= M_NODES * NOFF) return;
  int i = tid / NOFF, t = tid - i * NOFF;
  int xi = i >> 6, yi = i & 63;
  int xj = xi + OFF_DR[t], yj = yi + OFF_DC[t];
  float sc = 0.0f;
  if (xj >= 0 && xj < 64 && yj >= 0 && yj < 64) {
    float p1[9], p2[9];
    for (int a = 0; a < 3; ++a)
      for (int b = 0; b < 3; ++b) {
        p1[a * 3 + b] = imval(img, yi + a - 1, xi + b - 1);
        p2[a * 3 + b] = imval(img, yj + a - 1, xi + b - 1);
      }
    float s1 = lenet9(p1, cw, cb, f1w, f1b, f2w, f2b, f3w, f3b);
    float s2 = lenet9(p2, cw, cb, f1w, f1b, f2w, f2b, f3w, f3b);
    sc = 1.0f / (1.0f + expf(-(s1 * s2)));
  }
  Sdir[i * 16 + t] = sc;
}

// --------------------------------------------------- symmetrize + laplacian -
__global__ __launch_bounds__(256) void k_symdeg(const float* __restrict__ Sdir,
                                                float* __restrict__ Wb,
                                                float* __restrict__ deg) {
  int i = blockIdx.x * 256 + threadIdx.x;
  if (i >= M_NODES) return;
  int xi = i >> 6, yi = i & 63;
  float dv = 0.0f;
  for (int t = 0; t < NOFF; ++t) {
    int xj = xi + OFF_DR[t], yj = yi + OFF_DC[t];
    float wv = 0.0f;
    if (xj >= 0 && xj < 64 && yj >= 0 && yj < 64) {
      int j = (xj << 6) + yj;
      wv = 0.5f * (Sdir[i * 16 + t] + Sdir[j * 16 + (12 - t)]);
    }
    Wb[i * 16 + t] = wv;
    dv += wv;
  }
  Wb[i * 16 + 13] = 0.0f; Wb[i * 16 + 14] = 0.0f; Wb[i * 16 + 15] = 0.0f;
  deg[i] = dv;
}

__global__ __launch_bounds__(256) void k_lap(const float* __restrict__ Wb,
                                             const float* __restrict__ deg,
                                             float* __restrict__ Lb) {
  int i = blockIdx.x * 256 + threadIdx.x;
  if (i >= M_NODES) return;
  int xi = i >> 6, yi = i & 63;
  float di = deg[i];
  for (int t = 0; t < 16; ++t) {
    float lv = 0.0f;
    if (t < NOFF) {
      int xj = xi + OFF_DR[t], yj = yi + OFF_DC[t];
      if (xj >= 0 && xj < 64 && yj >= 0 && yj < 64) {
        int j = (xj << 6) + yj;
        lv = Wb[i * 16 + t] * rsqrtf(di * deg[j]);   // L = D^-1/2 A D^-1/2
      }
    }
    Lb[i * 16 + t] = lv;
  }
}

// ------------------------------------------------------------ Lanczos core --
__device__ __forceinline__ double blkred(double v, double* red) {
  int t = threadIdx.x;
  red[t] = v;
  __syncthreads();
  for (int off = 512; off > 0; off >>= 1) {
    if (t < off) red[t] += red[t + off];
    __syncthreads();
  }
  double r = red[0];
  __syncthreads();
  return r;
}

__global__ __launch_bounds__(1024) void k_lanczos16(
    const float* __restrict__ Lb, float* __restrict__ Qc,
    double* __restrict__ ad, double* __restrict__ bd, int k0) {
  __shared__ float qc[M_NODES];
  __shared__ float qp[M_NODES];
  __shared__ float vv[M_NODES];
  __shared__ double red[1024];
  const int t = threadIdx.x;

  if (k0 == 0) {
    for (int i = t; i < M_NODES; i += 1024) { qc[i] = (i == 1) ? 1.0f : 0.0f; qp[i] = 0.0f; }
  } else {
    for (int i = t; i < M_NODES; i += 1024) {
      qc[i] = Qc[(size_t)(k0 - 1) * M_NODES + i];
      qp[i] = Qc[(size_t)(k0 - 2) * M_NODES + i];
    }
  }
  __syncthreads();
  double beta = (k0 == 0) ? 0.0 : bd[k0];

  for (int s = 0; s < BLK; ++s) {
    const int k = k0 + s;
    // banded matvec vv = L * qc (13-point stencil, qc in LDS)
    for (int i = t; i < M_NODES; i += 1024) {
      const int r = i >> 6, c = i & 63;
      float acc = 0.0f;
#pragma unroll
      for (int tt = 0; tt < NOFF; ++tt) {
        int rr = r + OFF_DR[tt], cc = c + OFF_DC[tt];
        bool ok = (rr >= 0) && (rr < 64) && (cc >= 0) && (cc < 64);
        int j = ok ? (rr * 64 + cc) : i;
        acc += ok ? Lb[i * 16 + tt] * qc[j] : 0.0f;
      }
      vv[i] = acc;
    }
    __syncthreads();
    double loc = 0.0;
    for (int i = t; i < M_NODES; i += 1024) loc += (double)qc[i] * (double)vv[i];
    double alpha = blkred(loc, red);
    {
      float fa = (float)alpha, fb = (float)beta;
      for (int i = t; i < M_NODES; i += 1024) vv[i] -= fa * qc[i] + fb * qp[i];
    }
    __syncthreads();
    // second local orthogonalization pass against qc/qp
    double l1 = 0.0, l2 = 0.0;
    for (int i = t; i < M_NODES; i += 1024) { l1 += (double)vv[i] * qc[i]; l2 += (double)vv[i] * qp[i]; }
    double d1 = blkred(l1, red);
    double d2 = blkred(l2, red);
    {
      float f1 = (float)d1, f2 = (float)d2;
      for (int i = t; i < M_NODES; i += 1024) vv[i] -= f1 * qc[i] + f2 * qp[i];
    }
    alpha += d1;
    __syncthreads();
    loc = 0.0;
    for (int i = t; i < M_NODES; i += 1024) loc += (double)vv[i] * (double)vv[i];
    double nb = sqrt(blkred(loc, red));
    double betaT = nb;
    if (nb < 1e-10) {                       // breakdown -> deterministic restart
      for (int i = t; i < M_NODES; i += 1024) {
        unsigned h = (unsigned)(i * 2654435761u) ^ (unsigned)(k * 40503u + 2463534242u);
        h ^= h >> 13; h *= 2246822519u; h ^= h >> 16;
        vv[i] = (float)((double)(h & 0xFFFFFFu) * (1.0 / 8388608.0) - 1.0);
      }
      __syncthreads();
      l1 = 0.0; l2 = 0.0;
      for (int i = t; i < M_NODES; i += 1024) { l1 += (double)vv[i] * qc[i]; l2 += (double)vv[i] * qp[i]; }
      double e1 = blkred(l1, red), e2 = blkred(l2, red);
      float f1 = (float)e1, f2 = (float)e2;
      for (int i = t; i < M_NODES; i += 1024) vv[i] -= f1 * qc[i] + f2 * qp[i];
      __syncthreads();
      loc = 0.0;
      for (int i = t; i < M_NODES; i += 1024) loc += (double)vv[i] * (double)vv[i];
      nb = sqrt(blkred(loc, red));
      betaT = 0.0;                          // T becomes block-diagonal here
    }
    if (t == 0) { ad[k] = alpha; bd[k + 1] = betaT; }
    const float inv = (float)(1.0 / nb);
    for (int i = t; i < M_NODES; i += 1024) {
      float nv = vv[i] * inv;
      qp[i] = qc[i];
      qc[i] = nv;
      Qc[(size_t)k * M_NODES + i] = nv;     // column-major Q
    }
    __syncthreads();
    beta = betaT;
  }
}

// ----------------------------------------- WMMA block reorth: C = Q_old^T V -
__global__ __launch_bounds__(128) void k_project(const float* __restrict__ Qc,
                                                 float* __restrict__ Cg, int k0) {
  __shared__ float sacc[4][32][8];
  const int p0 = blockIdx.x * 16;           // old-column tile
  const int wave = threadIdx.x >> 5;
  const int lane = threadIdx.x & 31;
  const int h = lane >> 4, lm = lane & 15;
  const float* Acol = Qc + (size_t)(p0 + lm) * M_NODES;
  const float* Bcol = Qc + (size_t)(k0 + lm) * M_NODES;
  v8f acc = {};
  const int i0beg = wave * 1024, i0end = i0beg + 1024;   // K split across 4 waves
  for (int i0 = i0beg; i0 < i0end; i0 += 4) {
    const int ia = i0 + 2 * h;
    if ((i0 & 63) == 0) {
      __builtin_prefetch(Acol + ia + 256, 0, 1);
      __builtin_prefetch(Bcol + ia + 256, 0, 1);
    }
    v2f a, b;
    a.x = Acol[ia]; a.y = Acol[ia + 1];     // A: 16x4 tile of Q_old^T
    b.x = Bcol[ia]; b.y = Bcol[ia + 1];     // B: 4x16 tile of V
    acc = __builtin_amdgcn_wmma_f32_16x16x4_f32(false, a, false, b, (short)0, acc, false, false);
  }
#pragma unroll
  for (int r = 0; r < 8; ++r) sacc[wave][lane][r] = acc[r];
  __syncthreads();
  if (threadIdx.x < 32) {
    const int l = threadIdx.x, hh = l >> 4, ll = l & 15;
#pragma unroll
    for (int r = 0; r < 8; ++r) {
      float sres = sacc[0][l][r] + sacc[1][l][r] + sacc[2][l][r] + sacc[3][l][r];
      Cg[(p0 + r + 8 * hh) * 16 + ll] = sres;   // C[p][n], row-major 16-wide
    }
  }
}

// ------------------------------------------- WMMA apply: V -= Q_old * C -----
__global__ __launch_bounds__(128) void k_apply(float* __restrict__ Qc,
                                               const float* __restrict__ Cg, int k0) {
  __shared__ float sacc[4][32][8];
  const int i0 = blockIdx.x * 16;           // row tile of V
  const int wave = threadIdx.x >> 5;
  const int lane = threadIdx.x & 31;
  const int h = lane >> 4, lm = lane & 15;
  v8f acc = {};
  const int cpw = k0 >> 4;                  // K-chunks (of 4 p) per wave
  int pp = wave * (cpw << 2) + 2 * h;
  for (int c = 0; c < cpw; ++c, pp += 4) {
    v2f a, b;
    a.x = Qc[(size_t)pp * M_NODES + i0 + lm];          // A: 16x4 tile of Q_old
    a.y = Qc[(size_t)(pp + 1) * M_NODES + i0 + lm];
    b.x = Cg[pp * 16 + lm];                            // B: 4x16 tile of C
    b.y = Cg[(pp + 1) * 16 + lm];
    acc = __builtin_amdgcn_wmma_f32_16x16x4_f32(false, a, false, b, (short)0, acc, false, false);
  }
#pragma unroll
  for (int r = 0; r < 8; ++r) sacc[wave][lane][r] = acc[r];
  __syncthreads();
  if (threadIdx.x < 32) {
    const int l = threadIdx.x, hh = l >> 4, ll = l & 15;
#pragma unroll
    for (int r = 0; r < 8; ++r) {
      float sres = sacc[0][l][r] + sacc[1][l][r] + sacc[2][l][r] + sacc[3][l][r];
      size_t idx = (size_t)(k0 + ll) * M_NODES + (size_t)(i0 + r + 8 * hh);
      Qc[idx] = Qc[idx] - sres;
    }
  }
}

__global__ __launch_bounds__(256) void k_renorm(float* __restrict__ Qc, int k0) {
  __shared__ double red[256];
  float* base = Qc + (size_t)(k0 + blockIdx.x) * M_NODES;
  const int t = threadIdx.x;
  double loc = 0.0;
  for (int i = t; i < M_NODES; i += 256) { double x = base[i]; loc += x * x; }
  red[t] = loc; __syncthreads();
  for (int off = 128; off > 0; off >>= 1) { if (t < off) red[t] += red[t + off]; __syncthreads(); }
  double nrm = sqrt(red[0]);
  float inv = (nrm > 1e-30) ? (float)(1.0 / nrm) : 0.0f;
  for (int i = t; i < M_NODES; i += 256) base[i] *= inv;
}

// ------------------------------- tridiagonal eigenvalues (Sturm bisection) --
__global__ __launch_bounds__(256) void k_eigs(const double* __restrict__ ad,
                                              const double* __restrict__ bd,
                                              double* __restrict__ lam) {
  int k = blockIdx.x * 256 + threadIdx.x;
  if (k >= M_NODES) return;
  double lo = -2.0, hi = 2.0;               // spectrum of norm. adjacency in [-1,1]
  for (int it = 0; it < 44; ++it) {
    double x = 0.5 * (lo + hi);
    int cnt = 0;
    double d = 1.0;
    for (int m = 0; m < M_NODES; ++m) {
      double bm = (m > 0) ? bd[m] : 0.0;
      d = ad[m] - x - bm * bm / d;
      if (fabs(d) < 1e-300) d = -1e-300;
      if (d < 0.0) cnt++;
    }
    if (cnt > k) hi = x; else lo = x;
  }
  lam[k] = 0.5 * (lo + hi);
}

// -------------------- Golub-Welsch first components: w_k = 1/sqrt(sum p_m^2) -
__global__ __launch_bounds__(256) void k_weights(const double* __restrict__ ad,
                                                 const double* __restrict__ bd,
                                                 const double* __restrict__ lam,
                                                 double* __restrict__ wgt) {
  int k = blockIdx.x * 256 + threadIdx.x;
  if (k >= M_NODES) return;
  double x = lam[k];
  double pm = 1.0, pmm = 0.0, s = 1.0;
  bool dead = false;
  for (int m = 0; m < M_NODES - 1; ++m) {
    double bn = bd[m + 1];
    if (bn < 1e-12) break;                  // block boundary (restart)
    double bp = (m > 0) ? bd[m] : 0.0;
    double pn = ((x - ad[m]) * pm - bp * pmm) / bn;
    s += pn * pn;
    pmm = pm; pm = pn;
    if (!(s < 1e120)) { dead = true; break; }   // catches overflow and NaN
  }
  wgt[k] = dead ? 0.0 : 1.0 / sqrt(s);
}

__global__ __launch_bounds__(256) void k_out(const double* __restrict__ wgt,
                                             float* __restrict__ out) {
  int k = blockIdx.x * 256 + threadIdx.x;
  if (k >= M_NODES) return;
  double s0 = wgt[0];
  double sg = (s0 > 0.0) ? 1.0 : ((s0 < 0.0) ? -1.0 : 0.0);
  out[k] = (float)(wgt[k] * sg);
}

// ---------------------------------------------------------------- driver ----
extern "C" void kernel_launch(void* const* d_in, const int* in_sizes, int n_in,
                              void* d_out, int out_size, void* d_ws, size_t ws_size,
                              hipStream_t stream) {
  (void)in_sizes; (void)n_in; (void)out_size; (void)ws_size;
  const float* img = (const float*)d_in[0];
  const float* cw  = (const float*)d_in[1];
  const float* cb  = (const float*)d_in[2];
  const float* f1w = (const float*)d_in[3];
  const float* f1b = (const float*)d_in[4];
  const float* f2w = (const float*)d_in[5];
  const float* f2b = (const float*)d_in[6];
  const float* f3w = (const float*)d_in[7];
  const float* f3b = (const float*)d_in[8];
  float* out = (float*)d_out;

  char* ws = (char*)d_ws;
  float*  Qc   = (float*)(ws);                                        // 64 MB, column-major Q
  float*  Sdir = (float*)(ws + 67108864);                             // 256 KB
  float*  Wb   = (float*)(ws + 67108864 + 1 * 262144);                // 256 KB
  float*  Lb   = (float*)(ws + 67108864 + 2 * 262144);                // 256 KB
  float*  deg  = (float*)(ws + 67108864 + 3 * 262144);                // 16 KB
  float*  Cg   = (float*)(ws + 67108864 + 3 * 262144 + 16384);        // 256 KB
  double* ad   = (double*)(ws + 67108864 + 4 * 262144 + 16384);       // 32 KB
  double* bd   = ad + 4096;                                           // 4097 used
  double* lam  = bd + 4104;
  double* wgt  = lam + 4096;

  k_scores<<<dim3((M_NODES * NOFF + 255) / 256), dim3(256), 0, stream>>>(
      img, cw, cb, f1w, f1b, f2w, f2b, f3w, f3b, Sdir);
  k_symdeg<<<dim3(16), dim3(256), 0, stream>>>(Sdir, Wb, deg);
  k_lap<<<dim3(16), dim3(256), 0, stream>>>(Wb, deg, Lb);

  for (int kb = 0; kb < NBLK; ++kb) {
    int k0 = kb * BLK;
    k_lanczos16<<<dim3(1), dim3(1024), 0, stream>>>(Lb, Qc, ad, bd, k0);
    if (k0 > 0) {
      k_project<<<dim3(kb), dim3(128), 0, stream>>>(Qc, Cg, k0);
      k_apply<<<dim3(256), dim3(128), 0, stream>>>(Qc, Cg, k0);
      k_renorm<<<dim3(16), dim3(256), 0, stream>>>(Qc, k0);
    }
  }

  k_eigs<<<dim3(16), dim3(256), 0, stream>>>(ad, bd, lam);
  k_weights<<<dim3(16), dim3(256), 0, stream>>>(ad, bd, lam, wgt);
  k_out<<<dim3(16), dim3(256), 0, stream>>>(wgt, out);
}